// YOLOLikeLoss_24489903522532
// MI455X (gfx1250) — compile-verified
//
#include <hip/hip_runtime.h>
#include <stdint.h>

// Problem constants (match reference: B=64, P=8400, T=100)
#define BATCH   64
#define P_PREDS 8400
#define T_TGTS  100
#define SLICES  4
#define T_SLICE (T_TGTS / SLICES)   // 25 targets per block
#define BLOCK   512
#define WAVES   (BLOCK / 32)

// LDS layout (bytes)
#define OFF_CORN 0                              // float4[P]  corners (x1,y1,x2,y2)
#define OFF_AREA (P_PREDS * 16)                 // float[P]   pred area
#define OFF_TGT  (OFF_AREA + P_PREDS * 4)       // float[25*5] raw target records
#define OFF_PART (OFF_TGT + T_SLICE * 5 * 4)    // float[WAVES] per-wave partials
#define SMEM_BYTES ((OFF_PART + WAVES * 4 + 15) & ~15)

#define AS1 __attribute__((address_space(1)))
#define AS3 __attribute__((address_space(3)))

__global__ __launch_bounds__(BLOCK, 1)
void yolo_match_kernel(const float* __restrict__ pred,     // [B,P,5]
                       const float* __restrict__ tgt,      // [B,T,5]
                       const int*   __restrict__ lengths,  // [B]
                       float* __restrict__ partial)        // [B*SLICES]
{
    extern __shared__ char smem[];
    float4* sc = (float4*)(smem + OFF_CORN);
    float*  sa = (float*) (smem + OFF_AREA);
    float*  st = (float*) (smem + OFF_TGT);
    float*  sp = (float*) (smem + OFF_PART);

    const int b    = blockIdx.x / SLICES;
    const int s    = blockIdx.x % SLICES;
    const int tid  = threadIdx.x;
    const int lane = tid & 31;
    const int wave = tid >> 5;

    const float* predB = pred + (size_t)b * P_PREDS * 5;
    const float* tgtS  = tgt + ((size_t)b * T_TGTS + (size_t)s * T_SLICE) * 5;

    // ---- Stage targets (25 records x 5 floats = 125 dwords) to LDS via the
    //      CDNA5 async global->LDS path (tracked by ASYNCcnt), overlapped with
    //      the prediction preprocessing below.
    if (tid < T_SLICE * 5) {
#if __has_builtin(__builtin_amdgcn_global_load_async_to_lds_b32)
        __builtin_amdgcn_global_load_async_to_lds_b32(
            (AS1 int*)(uintptr_t)(tgtS + tid),
            (AS3 int*)(uintptr_t)(st + tid),
            0, 0);
#else
        st[tid] = tgtS[tid];
#endif
    }

    // ---- Preprocess predictions into LDS: corners + area (168 KB total).
    for (int i = tid; i < P_PREDS; i += BLOCK) {
        const float cx = predB[i * 5 + 1];
        const float cy = predB[i * 5 + 2];
        const float w  = predB[i * 5 + 3];
        const float h  = predB[i * 5 + 4];
        const float hw = 0.5f * w, hh = 0.5f * h;
        float4 c;
        c.x = cx - hw; c.y = cy - hh; c.z = cx + hw; c.w = cy + hh;
        sc[i] = c;
        sa[i] = (c.z - c.x) * (c.w - c.y);  // same association as reference
    }

#if __has_builtin(__builtin_amdgcn_s_wait_asynccnt)
    __builtin_amdgcn_s_wait_asynccnt(0);
#else
    asm volatile("s_wait_asynccnt 0" ::: "memory");
#endif
    __syncthreads();

    const int len = lengths[b];
    float wave_acc = 0.0f;  // accumulated by lane 0 of each wave, fixed order

    // ---- One target per wave; lanes stride the 8400 predictions.
    for (int tw = wave; tw < T_SLICE; tw += WAVES) {
        const int t = s * T_SLICE + tw;
        if (t >= len) continue;  // masked target: contributes 0, skip the scan

        const float tcx = st[tw * 5 + 1], tcy = st[tw * 5 + 2];
        const float twd = st[tw * 5 + 3], thd = st[tw * 5 + 4];
        const float tx1 = tcx - 0.5f * twd, ty1 = tcy - 0.5f * thd;
        const float tx2 = tcx + 0.5f * twd, ty2 = tcy + 0.5f * thd;
        const float tarea = (tx2 - tx1) * (ty2 - ty1);

        // Rational argmax state: iou = bnum/bden, bden > 0. Init so the first
        // candidate always wins (matches argmax first-occurrence semantics).
        float bnum = -1.0f, bden = 1.0f;
        int   bidx = 0;

        for (int i = lane; i < P_PREDS; i += 32) {
            const float4 c = sc[i];
            const float ax = fmaxf(c.x, tx1);
            const float ay = fmaxf(c.y, ty1);
            const float bx = fminf(c.z, tx2);
            const float by = fminf(c.w, ty2);
            const float iw = fmaxf(bx - ax, 0.0f);
            const float ih = fmaxf(by - ay, 0.0f);
            const float inter = iw * ih;
            // same association as reference: ((area_p + area_t) - inter) + 1e-6
            const float den = sa[i] + tarea - inter + 1e-6f;
            // inter/den > bnum/bden  <=>  inter*bden > bnum*den (both dens > 0)
            if (inter * bden > bnum * den) { bnum = inter; bden = den; bidx = i; }
        }

        // Wave32 butterfly reduce: max value, ties -> smallest index.
        #pragma unroll
        for (int off = 16; off > 0; off >>= 1) {
            const float on = __shfl_xor(bnum, off, 32);
            const float od = __shfl_xor(bden, off, 32);
            const int   oi = __shfl_xor(bidx, off, 32);
            const float l = on * bden, r = bnum * od;
            if (l > r || (l == r && oi < bidx)) { bnum = on; bden = od; bidx = oi; }
        }

        if (lane == 0) {
            const float* mp = predB + (size_t)bidx * 5;
            float sq = 0.0f;
            #pragma unroll
            for (int k = 0; k < 5; ++k) {
                const float d = mp[k] - st[tw * 5 + k];
                sq += d * d;
            }
            wave_acc += sq;
        }
    }

    if (lane == 0) sp[wave] = wave_acc;
    __syncthreads();

    if (tid == 0) {
        float acc = 0.0f;
        #pragma unroll
        for (int wv = 0; wv < WAVES; ++wv) acc += sp[wv];
        partial[blockIdx.x] = acc;
    }
}

// Deterministic fixed-tree reduction of the 256 per-block partials.
__global__ __launch_bounds__(BATCH * SLICES, 1)
void yolo_reduce_kernel(const float* __restrict__ partial, float* __restrict__ out)
{
    __shared__ float sm[BATCH * SLICES];
    const int tid = threadIdx.x;
    sm[tid] = partial[tid];
    __syncthreads();
    for (int sdis = (BATCH * SLICES) / 2; sdis > 0; sdis >>= 1) {
        if (tid < sdis) sm[tid] += sm[tid + sdis];
        __syncthreads();
    }
    if (tid == 0) out[0] = sm[0] * (1.0f / (float)BATCH);
}

extern "C" void kernel_launch(void* const* d_in, const int* in_sizes, int n_in,
                              void* d_out, int out_size, void* d_ws, size_t ws_size,
                              hipStream_t stream) {
    const float* pred    = (const float*)d_in[0];  // [64, 8400, 5] f32
    const float* tgt     = (const float*)d_in[1];  // [64, 100, 5]  f32
    const int*   lengths = (const int*)  d_in[2];  // [64] (int32 on device)
    float* out     = (float*)d_out;                // scalar loss
    float* partial = (float*)d_ws;                 // 256 floats of scratch

    (void)in_sizes; (void)n_in; (void)out_size; (void)ws_size;

    // 168+ KB dynamic LDS: opt in explicitly (no-op if already allowed).
    (void)hipFuncSetAttribute((const void*)yolo_match_kernel,
                              hipFuncAttributeMaxDynamicSharedMemorySize,
                              SMEM_BYTES);

    yolo_match_kernel<<<BATCH * SLICES, BLOCK, SMEM_BYTES, stream>>>(
        pred, tgt, lengths, partial);
    yolo_reduce_kernel<<<1, BATCH * SLICES, 0, stream>>>(partial, out);
}